// Generator_67250597921369
// MI455X (gfx1250) — compile-verified
//
#include <hip/hip_runtime.h>
#include <hip/hip_bf16.h>

typedef float v2f __attribute__((ext_vector_type(2)));
typedef float v8f __attribute__((ext_vector_type(8)));

#define WMMA_F32(a, b, c) \
  __builtin_amdgcn_wmma_f32_16x16x4_f32(false, (a), false, (b), (short)0, (c), false, false)

constexpr int BATCH   = 4;
constexpr int SEQLEN  = 4096;
constexpr int DMODEL  = 256;
constexpr int DSTATE  = 64;
constexpr int DCONV   = 4;
constexpr int HEADDIM = 64;
constexpr int CHUNK   = 64;
constexpr int DINNER  = 1024;           // EXPAND * DMODEL
constexpr int NHEADS  = 16;             // DINNER / HEADDIM
constexpr int CONVDIM = 1152;           // DINNER + 2*DSTATE
constexpr int DINPROJ = 2192;           // 2*DINNER + 2*DSTATE + NHEADS
constexpr int NCHUNK  = SEQLEN / CHUNK; // 64

__device__ __forceinline__ float silu_f(float x) { return x / (1.0f + expf(-x)); }

// ---------------------------------------------------------------------------
// K1: in_proj GEMM  zx[m, j] = sum_k u[m,k] * W[j,k]   (M=16384, N=2192, K=256)
// Wave computes 4 consecutive 16x16 M-tiles sharing one B fragment per k-step.
// ---------------------------------------------------------------------------
__global__ __launch_bounds__(128) void k_in_proj(const float* __restrict__ U,
                                                 const float* __restrict__ W,
                                                 float* __restrict__ out) {
  int lane = threadIdx.x & 31, wave = threadIdx.x >> 5;
  int mtBase = (blockIdx.y * 4 + wave) * 4;  // 4 M-tiles per wave, 0..1020
  int nt = blockIdx.x;                       // 0..136
  int half = lane >> 4, lm = lane & 15;
  const float* a0 = U + (size_t)((mtBase + 0) * 16 + lm) * DMODEL + 2 * half;
  const float* a1 = U + (size_t)((mtBase + 1) * 16 + lm) * DMODEL + 2 * half;
  const float* a2 = U + (size_t)((mtBase + 2) * 16 + lm) * DMODEL + 2 * half;
  const float* a3 = U + (size_t)((mtBase + 3) * 16 + lm) * DMODEL + 2 * half;
  const float* brow = W + (size_t)(nt * 16 + lm) * DMODEL + 2 * half;
  v8f acc0 = {0.f,0.f,0.f,0.f,0.f,0.f,0.f,0.f};
  v8f acc1 = acc0, acc2 = acc0, acc3 = acc0;
  for (int k = 0; k < DMODEL; k += 4) {
    v2f b = *(const v2f*)(brow + k);
    acc0 = WMMA_F32(*(const v2f*)(a0 + k), b, acc0);
    acc1 = WMMA_F32(*(const v2f*)(a1 + k), b, acc1);
    acc2 = WMMA_F32(*(const v2f*)(a2 + k), b, acc2);
    acc3 = WMMA_F32(*(const v2f*)(a3 + k), b, acc3);
  }
#pragma unroll
  for (int i = 0; i < 4; i++) {
    v8f acc = (i == 0) ? acc0 : (i == 1) ? acc1 : (i == 2) ? acc2 : acc3;
    float* o = out + (size_t)((mtBase + i) * 16 + 8 * half) * DINPROJ + nt * 16 + lm;
#pragma unroll
    for (int r = 0; r < 8; r++) o[(size_t)r * DINPROJ] = acc[r];
  }
}

// ---------------------------------------------------------------------------
// K2a: depthwise causal conv (width 4) + bias + SiLU over xBC channels
// ---------------------------------------------------------------------------
__global__ __launch_bounds__(256) void k_conv(const float* __restrict__ zx,
                                              const float* __restrict__ cw,
                                              const float* __restrict__ cb,
                                              float* __restrict__ xBCc) {
  size_t i = (size_t)blockIdx.x * 256 + threadIdx.x;
  if (i >= (size_t)BATCH * SEQLEN * CONVDIM) return;
  int ch = (int)(i % CONVDIM);
  size_t bl = i / CONVDIM;
  int l = (int)(bl % SEQLEN);
  size_t b = bl / SEQLEN;
  float acc = cb[ch];
#pragma unroll
  for (int w = 0; w < DCONV; w++) {
    int ls = l - (DCONV - 1) + w;
    if (ls >= 0) acc += zx[((size_t)b * SEQLEN + ls) * DINPROJ + DINNER + ch] * cw[ch * DCONV + w];
  }
  xBCc[i] = silu_f(acc);
}

// ---------------------------------------------------------------------------
// K2b: dt = softplus(dt_raw + dt_bias); dA = -exp(A_log) * dt
// ---------------------------------------------------------------------------
__global__ __launch_bounds__(256) void k_dt(const float* __restrict__ zx,
                                            const float* __restrict__ dt_bias,
                                            const float* __restrict__ A_log,
                                            float* __restrict__ dt, float* __restrict__ dA) {
  size_t i = (size_t)blockIdx.x * 256 + threadIdx.x;
  if (i >= (size_t)BATCH * SEQLEN * NHEADS) return;
  int h = (int)(i % NHEADS);
  size_t bl = i / NHEADS;
  float x = zx[bl * DINPROJ + (DINPROJ - NHEADS) + h] + dt_bias[h];
  float sp = (x > 20.f) ? x : log1pf(expf(x));
  dt[i] = sp;
  dA[i] = -expf(A_log[h]) * sp;
}

// ---------------------------------------------------------------------------
// K3: per-chunk kernel: G = C @ B^T, Y_diag = (G .* L_h) @ xdt_h,
//     states_h = (xdt .* decay)^T @ B.  All WMMA f32 16x16x4.
//     L factored with mid-chunk centering: L[m,k] = expA[m] * expnA[k].
//     Cx is Cs during the G phase, then re-used per head as the staged
//     pre-multiplied xdt tile (xdt * dts * expnA).
// ---------------------------------------------------------------------------
__global__ __launch_bounds__(256) void k_chunk(const float* __restrict__ xBCc,
                                               const float* __restrict__ dt,
                                               const float* __restrict__ dA,
                                               float* __restrict__ states,
                                               float* __restrict__ Alast,
                                               float* __restrict__ y) {
  __shared__ float Bs[CHUNK][DSTATE + 1];
  __shared__ float Cx[CHUNK][DSTATE + 1];   // C matrix, later staged xdt tile
  __shared__ float Gs[CHUNK][CHUNK + 1];
  __shared__ float expA[NHEADS][CHUNK];     // exp(cs - ref)
  __shared__ float expnA[NHEADS][CHUNK];    // exp(ref - cs)
  __shared__ float dts[NHEADS][CHUNK];
  __shared__ float refA[NHEADS];

  int bc = blockIdx.x;
  int c = bc % NCHUNK, b = bc / NCHUNK;
  int tid = threadIdx.x;
  size_t l0 = (size_t)b * SEQLEN + (size_t)c * CHUNK;

  for (int idx = tid; idx < CHUNK * DSTATE; idx += 256) {
    int l = idx >> 6, n = idx & 63;
    Bs[l][n] = xBCc[(l0 + l) * CONVDIM + DINNER + n];
    Cx[l][n] = xBCc[(l0 + l) * CONVDIM + DINNER + DSTATE + n];
  }
  for (int idx = tid; idx < NHEADS * CHUNK; idx += 256) {
    int h = idx >> 6, l = idx & 63;
    dts[h][l] = dt[(l0 + l) * NHEADS + h];
    expA[h][l] = dA[(l0 + l) * NHEADS + h];  // raw dA for now
  }
  __syncthreads();
  if (tid < NHEADS) {
    float cs = 0.f;
    for (int l = 0; l < CHUNK; l++) { cs += expA[tid][l]; expA[tid][l] = cs; }
    refA[tid] = expA[tid][31];
    Alast[(size_t)bc * NHEADS + tid] = cs;  // uncentered chunk-end cumsum
  }
  __syncthreads();
  for (int idx = tid; idx < NHEADS * CHUNK; idx += 256) {
    int h = idx >> 6, l = idx & 63;
    float cs = expA[h][l], r = refA[h];
    expnA[h][l] = expf(r - cs);
    expA[h][l] = expf(cs - r);
  }
  __syncthreads();

  int lane = tid & 31, wave = tid >> 5;
  int half = lane >> 4, lm = lane & 15;

  // G[l,s] = sum_n C[l,n] * B[s,n]
  for (int t = wave; t < 16; t += 8) {
    int mt = t >> 2, st = t & 3;
    v8f acc = {0.f,0.f,0.f,0.f,0.f,0.f,0.f,0.f};
    for (int k = 0; k < DSTATE; k += 4) {
      int kk = k + 2 * half;
      v2f a, bf;
      a.x = Cx[mt * 16 + lm][kk];  a.y = Cx[mt * 16 + lm][kk + 1];
      bf.x = Bs[st * 16 + lm][kk]; bf.y = Bs[st * 16 + lm][kk + 1];
      acc = WMMA_F32(a, bf, acc);
    }
#pragma unroll
    for (int r = 0; r < 8; r++) Gs[mt * 16 + 8 * half + r][st * 16 + lm] = acc[r];
  }
  __syncthreads();

  for (int h = 0; h < NHEADS; h++) {
    // Stage xdt_h * expnA into Cx (coalesced), replacing previous contents.
    __syncthreads();
    for (int idx = tid; idx < CHUNK * HEADDIM; idx += 256) {
      int l = idx >> 6, p = idx & 63;
      Cx[l][p] = xBCc[(l0 + l) * CONVDIM + h * HEADDIM + p] * dts[h][l] * expnA[h][l];
    }
    __syncthreads();
    float lastE = expA[h][CHUNK - 1];  // exp(cs_last - ref)
    for (int t = wave; t < 32; t += 8) {
      int tt = t & 15;
      int mt = tt >> 2, nt = tt & 3;
      v8f acc = {0.f,0.f,0.f,0.f,0.f,0.f,0.f,0.f};
      if (t < 16) {
        // Y_diag tile: A = mask(G), B[k,p] = Cx[k][p]; out scaled by expA[m]
        int m = mt * 16 + lm;
        for (int kt = 0; kt <= mt; kt++) {
          for (int k4 = 0; k4 < 16; k4 += 4) {
            int kg = kt * 16 + k4 + 2 * half;
            v2f a, bf;
            a.x = (m >= kg) ? Gs[m][kg] : 0.f;
            a.y = (m >= kg + 1) ? Gs[m][kg + 1] : 0.f;
            int col = nt * 16 + lm;
            bf.x = Cx[kg][col];
            bf.y = Cx[kg + 1][col];
            acc = WMMA_F32(a, bf, acc);
          }
        }
#pragma unroll
        for (int r = 0; r < 8; r++) {
          int mo = mt * 16 + 8 * half + r;
          y[(l0 + mo) * DINNER + h * HEADDIM + nt * 16 + lm] = acc[r] * expA[h][mo];
        }
      } else {
        // states tile: A[p,l] = Cx[l][p] * lastE, B[l,n] = Bs[l][n]
        int p = mt * 16 + lm;
        for (int kt = 0; kt < 4; kt++) {
          for (int k4 = 0; k4 < 16; k4 += 4) {
            int kg = kt * 16 + k4 + 2 * half;
            v2f a, bf;
            a.x = Cx[kg][p] * lastE;
            a.y = Cx[kg + 1][p] * lastE;
            bf.x = Bs[kg][nt * 16 + lm];
            bf.y = Bs[kg + 1][nt * 16 + lm];
            acc = WMMA_F32(a, bf, acc);
          }
        }
        float* sp = states + ((size_t)bc * NHEADS + h) * (HEADDIM * DSTATE);
#pragma unroll
        for (int r = 0; r < 8; r++)
          sp[(mt * 16 + 8 * half + r) * DSTATE + nt * 16 + lm] = acc[r];
      }
    }
  }
}

// ---------------------------------------------------------------------------
// K4: inter-chunk scan (in place: states[c] <- prefix state entering chunk c)
// ---------------------------------------------------------------------------
__global__ __launch_bounds__(256) void k_scan(float* __restrict__ states,
                                              const float* __restrict__ Alast) {
  int bh = blockIdx.x;
  int h = bh % NHEADS, b = bh / NHEADS;
  int tid = threadIdx.x;
  float run[16];
#pragma unroll
  for (int j = 0; j < 16; j++) run[j] = 0.f;
  for (int c = 0; c < NCHUNK; c++) {
    size_t bc = (size_t)b * NCHUNK + c;
    float* sp = states + (bc * NHEADS + h) * (HEADDIM * DSTATE);
    float ed = expf(Alast[bc * NHEADS + h]);
#pragma unroll
    for (int j = 0; j < 16; j++) {
      int e = j * 256 + tid;
      float s = sp[e];
      sp[e] = run[j];
      run[j] = run[j] * ed + s;
    }
  }
}

// ---------------------------------------------------------------------------
// K5: Y_off = exp(cs_l) * (C @ prev_h^T), plus D-skip; accumulate into y.
// ---------------------------------------------------------------------------
__global__ __launch_bounds__(256) void k_yoff(const float* __restrict__ xBCc,
                                              const float* __restrict__ dA,
                                              const float* __restrict__ prev,
                                              const float* __restrict__ Dp,
                                              float* __restrict__ y) {
  __shared__ float Cs[CHUNK][DSTATE + 1];
  __shared__ float Ps[HEADDIM][DSTATE + 1];
  __shared__ float eAcs[NHEADS][CHUNK];

  int bc = blockIdx.x;
  int c = bc % NCHUNK, b = bc / NCHUNK;
  int tid = threadIdx.x;
  size_t l0 = (size_t)b * SEQLEN + (size_t)c * CHUNK;

  for (int idx = tid; idx < CHUNK * DSTATE; idx += 256) {
    int l = idx >> 6, n = idx & 63;
    Cs[l][n] = xBCc[(l0 + l) * CONVDIM + DINNER + DSTATE + n];
  }
  for (int idx = tid; idx < NHEADS * CHUNK; idx += 256) {
    int h = idx >> 6, l = idx & 63;
    eAcs[h][l] = dA[(l0 + l) * NHEADS + h];
  }
  __syncthreads();
  if (tid < NHEADS) {
    float cs = 0.f;
    for (int l = 0; l < CHUNK; l++) { cs += eAcs[tid][l]; eAcs[tid][l] = cs; }
  }
  __syncthreads();
  for (int idx = tid; idx < NHEADS * CHUNK; idx += 256) {
    int h = idx >> 6, l = idx & 63;
    eAcs[h][l] = expf(eAcs[h][l]);
  }

  int lane = tid & 31, wave = tid >> 5;
  int half = lane >> 4, lm = lane & 15;

  for (int h = 0; h < NHEADS; h++) {
    __syncthreads();
    const float* sp = prev + ((size_t)bc * NHEADS + h) * (HEADDIM * DSTATE);
    for (int idx = tid; idx < HEADDIM * DSTATE; idx += 256)
      Ps[idx >> 6][idx & 63] = sp[idx];
    __syncthreads();
    float dskip = Dp[h];
    for (int t = wave; t < 16; t += 8) {
      int mt = t >> 2, nt = t & 3;
      v8f acc = {0.f,0.f,0.f,0.f,0.f,0.f,0.f,0.f};
      for (int k = 0; k < DSTATE; k += 4) {
        int kg = k + 2 * half;
        v2f a, bf;
        a.x = Cs[mt * 16 + lm][kg];   a.y = Cs[mt * 16 + lm][kg + 1];
        bf.x = Ps[nt * 16 + lm][kg];  bf.y = Ps[nt * 16 + lm][kg + 1];
        acc = WMMA_F32(a, bf, acc);
      }
#pragma unroll
      for (int r = 0; r < 8; r++) {
        int m = mt * 16 + 8 * half + r;
        size_t oi = (l0 + m) * DINNER + h * HEADDIM + nt * 16 + lm;
        float xv = xBCc[(l0 + m) * CONVDIM + h * HEADDIM + nt * 16 + lm];
        y[oi] += acc[r] * eAcs[h][m] + xv * dskip;
      }
    }
  }
}

// ---------------------------------------------------------------------------
// K6: y = rmsnorm(y * silu(z)) * norm_w, in place; one block per (b,l).
// ---------------------------------------------------------------------------
__global__ __launch_bounds__(256) void k_gatenorm(const float* __restrict__ zx,
                                                  const float* __restrict__ nw,
                                                  float* __restrict__ y) {
  __shared__ float red[8];
  __shared__ float srms;
  size_t bl = blockIdx.x;
  float vals[4];
  float ss = 0.f;
#pragma unroll
  for (int j = 0; j < 4; j++) {
    int idx = j * 256 + threadIdx.x;
    float z = zx[bl * DINPROJ + idx];
    float v = y[bl * DINNER + idx] * silu_f(z);
    vals[j] = v;
    ss += v * v;
  }
#pragma unroll
  for (int o = 16; o >= 1; o >>= 1) ss += __shfl_xor(ss, o, 32);
  if ((threadIdx.x & 31) == 0) red[threadIdx.x >> 5] = ss;
  __syncthreads();
  if (threadIdx.x == 0) {
    float t = 0.f;
    for (int w = 0; w < 8; w++) t += red[w];
    srms = rsqrtf(t / (float)DINNER + 1e-5f);
  }
  __syncthreads();
  float rms = srms;
#pragma unroll
  for (int j = 0; j < 4; j++) {
    int idx = j * 256 + threadIdx.x;
    y[bl * DINNER + idx] = vals[j] * rms * nw[idx];
  }
}

// ---------------------------------------------------------------------------
// K7: out_proj GEMM  out[m,d] = sum_j yn[m,j] * Wout[d,j]  (M=16384,N=256,K=1024)
// Same 4x M-blocking as K1.
// ---------------------------------------------------------------------------
__global__ __launch_bounds__(128) void k_out_proj(const float* __restrict__ Y,
                                                  const float* __restrict__ W,
                                                  float* __restrict__ out) {
  int lane = threadIdx.x & 31, wave = threadIdx.x >> 5;
  int mtBase = (blockIdx.y * 4 + wave) * 4;  // 0..1020
  int nt = blockIdx.x;                       // 0..15
  int half = lane >> 4, lm = lane & 15;
  const float* a0 = Y + (size_t)((mtBase + 0) * 16 + lm) * DINNER + 2 * half;
  const float* a1 = Y + (size_t)((mtBase + 1) * 16 + lm) * DINNER + 2 * half;
  const float* a2 = Y + (size_t)((mtBase + 2) * 16 + lm) * DINNER + 2 * half;
  const float* a3 = Y + (size_t)((mtBase + 3) * 16 + lm) * DINNER + 2 * half;
  const float* brow = W + (size_t)(nt * 16 + lm) * DINNER + 2 * half;
  v8f acc0 = {0.f,0.f,0.f,0.f,0.f,0.f,0.f,0.f};
  v8f acc1 = acc0, acc2 = acc0, acc3 = acc0;
  for (int k = 0; k < DINNER; k += 4) {
    v2f b = *(const v2f*)(brow + k);
    acc0 = WMMA_F32(*(const v2f*)(a0 + k), b, acc0);
    acc1 = WMMA_F32(*(const v2f*)(a1 + k), b, acc1);
    acc2 = WMMA_F32(*(const v2f*)(a2 + k), b, acc2);
    acc3 = WMMA_F32(*(const v2f*)(a3 + k), b, acc3);
  }
#pragma unroll
  for (int i = 0; i < 4; i++) {
    v8f acc = (i == 0) ? acc0 : (i == 1) ? acc1 : (i == 2) ? acc2 : acc3;
    float* o = out + (size_t)((mtBase + i) * 16 + 8 * half) * DMODEL + nt * 16 + lm;
#pragma unroll
    for (int r = 0; r < 8; r++) o[(size_t)r * DMODEL] = acc[r];
  }
}

// ---------------------------------------------------------------------------
extern "C" void kernel_launch(void* const* d_in, const int* in_sizes, int n_in,
                              void* d_out, int out_size, void* d_ws, size_t ws_size,
                              hipStream_t stream) {
  const float* u        = (const float*)d_in[0];
  const float* in_w     = (const float*)d_in[1];
  const float* conv_w   = (const float*)d_in[2];
  const float* conv_b   = (const float*)d_in[3];
  const float* dt_bias  = (const float*)d_in[4];
  const float* A_log    = (const float*)d_in[5];
  const float* Dp       = (const float*)d_in[6];
  const float* norm_w   = (const float*)d_in[7];
  const float* out_w    = (const float*)d_in[8];
  float* out = (float*)d_out;

  float* ws   = (float*)d_ws;
  float* zx   = ws;                                                 // B*L*DINPROJ
  float* xBCc = zx + (size_t)BATCH * SEQLEN * DINPROJ;              // B*L*CONVDIM
  float* dt   = xBCc + (size_t)BATCH * SEQLEN * CONVDIM;            // B*L*NHEADS
  float* dA   = dt + (size_t)BATCH * SEQLEN * NHEADS;               // B*L*NHEADS
  float* Alast= dA + (size_t)BATCH * SEQLEN * NHEADS;               // B*NCHUNK*NHEADS
  float* st   = Alast + (size_t)BATCH * NCHUNK * NHEADS;            // B*NCHUNK*NHEADS*64*64
  float* yb   = st + (size_t)BATCH * NCHUNK * NHEADS * HEADDIM * DSTATE;  // B*L*DINNER

  // K1: in_proj  (grid: 137 N-tiles x 64 groups of 16 M-tiles)
  hipLaunchKernelGGL(k_in_proj, dim3(DINPROJ / 16, BATCH * SEQLEN / 256), dim3(128), 0, stream,
                     u, in_w, zx);
  // K2: conv + silu; dt/dA
  {
    size_t n = (size_t)BATCH * SEQLEN * CONVDIM;
    hipLaunchKernelGGL(k_conv, dim3((n + 255) / 256), dim3(256), 0, stream, zx, conv_w, conv_b, xBCc);
  }
  {
    size_t n = (size_t)BATCH * SEQLEN * NHEADS;
    hipLaunchKernelGGL(k_dt, dim3((n + 255) / 256), dim3(256), 0, stream, zx, dt_bias, A_log, dt, dA);
  }
  // K3: per-chunk states + Y_diag
  hipLaunchKernelGGL(k_chunk, dim3(BATCH * NCHUNK), dim3(256), 0, stream,
                     xBCc, dt, dA, st, Alast, yb);
  // K4: inter-chunk scan (in place -> prefix states)
  hipLaunchKernelGGL(k_scan, dim3(BATCH * NHEADS), dim3(256), 0, stream, st, Alast);
  // K5: Y_off + D-skip
  hipLaunchKernelGGL(k_yoff, dim3(BATCH * NCHUNK), dim3(256), 0, stream,
                     xBCc, dA, st, Dp, yb);
  // K6: gate + RMSNorm (in place)
  hipLaunchKernelGGL(k_gatenorm, dim3(BATCH * SEQLEN), dim3(256), 0, stream, zx, norm_w, yb);
  // K7: out_proj
  hipLaunchKernelGGL(k_out_proj, dim3(DMODEL / 16, BATCH * SEQLEN / 256), dim3(128), 0, stream,
                     yb, out_w, out);
}